// MultiHeadedAttentionPooling_69441031242499
// MI455X (gfx1250) — compile-verified
//
#include <hip/hip_runtime.h>

typedef __attribute__((ext_vector_type(2))) float v2f;
typedef __attribute__((ext_vector_type(4))) float v4f;
typedef __attribute__((ext_vector_type(8))) float v8f;

// ---------------------------------------------------------------------------
// Stage 1: column-sum of Cp over each (n,h) block of 512 rows.
// Cp viewed as [256][512][512] fp32 (n*8+h outer). Streams 268 MB once, so it
// is the only HBM-bound stage (~11.5 us floor at 23.3 TB/s). Non-temporal
// loads keep the 268 MB stream from thrashing the 192 MB L2.
// ---------------------------------------------------------------------------
#define LSPLIT   8
#define SEG_ROWS (512 / LSPLIT)

__global__ void reduce_cp_partial(const float* __restrict__ Cp,
                                  float* __restrict__ partial) {
    const int blk = blockIdx.x;            // 0..255  == n*8+h
    const int seg = blockIdx.y;            // 0..LSPLIT-1
    const int t   = threadIdx.x;           // 0..127, 4 floats each (b128 loads)
    const float* base = Cp + (size_t)blk * (512 * 512)
                           + (size_t)seg * SEG_ROWS * 512 + t * 4;
    v4f acc = {0.f, 0.f, 0.f, 0.f};
    #pragma unroll 4
    for (int r = 0; r < SEG_ROWS; ++r) {
        const v4f v = __builtin_nontemporal_load((const v4f*)(base + (size_t)r * 512));
        acc += v;
    }
    *(v4f*)(partial + ((size_t)blk * LSPLIT + seg) * 512 + t * 4) = acc;
}

__global__ void reduce_cp_combine(const float* __restrict__ partial,
                                  float* __restrict__ Csum) {
    const int blk = blockIdx.x;            // 0..255
    const int t   = threadIdx.x;           // 0..127
    const float* p = partial + (size_t)blk * LSPLIT * 512 + t * 4;
    v4f acc = {0.f, 0.f, 0.f, 0.f};
    #pragma unroll
    for (int s = 0; s < LSPLIT; ++s)
        acc += *(const v4f*)(p + s * 512);
    *(v4f*)(Csum + (size_t)blk * 512 + t * 4) = acc;
}

// ---------------------------------------------------------------------------
// Generic fp32 WMMA GEMM:  C[M,N] = A[M,K] * B[N,K]^T  (+bias[N]) (relu) (+res)
// One wave (32 lanes) per 16x16 tile; K in steps of 4 via V_WMMA_F32_16X16X4_F32.
// A-frag (ISA 7.12.2, 32-bit A 16x4): lanes 0-15 -> M=lane, K = {k,k+1};
// lanes 16-31 -> M=lane-16, K = {k+2,k+3}. B-frag is the mirrored layout, so
// both operands are a per-lane contiguous float2 at offset k + (lane>>4)*2.
// D element j: row = m0 + j + (lane>>4)*8, col = n0 + (lane&15).
// ---------------------------------------------------------------------------
__global__ void wmma_gemm_nt_f32(const float* __restrict__ A,
                                 const float* __restrict__ B,
                                 const float* __restrict__ bias,
                                 const float* __restrict__ res,
                                 float* __restrict__ C,
                                 int M, int N, int K, int do_relu) {
    const int lane = threadIdx.x;          // 0..31 (wave32)
    const int hi   = lane >> 4;            // 0 or 1
    const int l15  = lane & 15;
    const int m0   = blockIdx.y * 16;
    const int n0   = blockIdx.x * 16;

    const float* arow = A + (size_t)(m0 + l15) * K;
    const float* brow = B + (size_t)(n0 + l15) * K;

    v8f acc = {0.f, 0.f, 0.f, 0.f, 0.f, 0.f, 0.f, 0.f};
    for (int k = 0; k < K; k += 4) {
        const int kb = k + hi * 2;
        const v2f a = *(const v2f*)(arow + kb);
        const v2f b = *(const v2f*)(brow + kb);
        acc = __builtin_amdgcn_wmma_f32_16x16x4_f32(
            /*neg_a=*/false, a, /*neg_b=*/false, b,
            /*c_mod=*/(short)0, acc, /*reuse_a=*/false, /*reuse_b=*/false);
    }

    const int   col = n0 + l15;
    const float bv  = bias ? bias[col] : 0.f;
    #pragma unroll
    for (int j = 0; j < 8; ++j) {
        const int row = m0 + j + hi * 8;
        float v = acc[j] + bv;
        if (do_relu) v = fmaxf(v, 0.f);
        if (res)     v += res[(size_t)row * N + col];
        C[(size_t)row * N + col] = v;
    }
}

// ---------------------------------------------------------------------------
// phi[n, h*64+d] = sum_{j=0..7} S[n*8+h, j*64+d]   (the reshape fold)
// ---------------------------------------------------------------------------
__global__ void fold_phi(const float* __restrict__ S, float* __restrict__ phi) {
    const int idx = blockIdx.x * blockDim.x + threadIdx.x;  // 0..16383
    const int n = idx >> 9;
    const int e = idx & 511;
    const int h = e >> 6;
    const int d = e & 63;
    const float* s = S + ((size_t)(n * 8 + h)) * 512 + d;
    float acc = 0.f;
    #pragma unroll
    for (int j = 0; j < 8; ++j) acc += s[j * 64];
    phi[idx] = acc;
}

// ---------------------------------------------------------------------------
// Row LayerNorm over E=512 (32 rows). 256 threads = 8 waves, shfl + LDS reduce.
// ---------------------------------------------------------------------------
__global__ void layernorm_rows(const float* __restrict__ x,
                               const float* __restrict__ gamma,
                               const float* __restrict__ beta,
                               float* __restrict__ out) {
    __shared__ float sbuf[8];
    __shared__ float qbuf[8];
    const int row = blockIdx.x;
    const int t   = threadIdx.x;           // 0..255
    const float* xr = x + (size_t)row * 512;
    const float a = xr[t];
    const float b = xr[t + 256];
    float s = a + b;
    float q = a * a + b * b;
    #pragma unroll
    for (int off = 16; off > 0; off >>= 1) {
        s += __shfl_down(s, off);
        q += __shfl_down(q, off);
    }
    const int wave = t >> 5;
    if ((t & 31) == 0) { sbuf[wave] = s; qbuf[wave] = q; }
    __syncthreads();
    if (t == 0) {
        float ts = 0.f, tq = 0.f;
        #pragma unroll
        for (int w = 0; w < 8; ++w) { ts += sbuf[w]; tq += qbuf[w]; }
        sbuf[0] = ts; qbuf[0] = tq;
    }
    __syncthreads();
    const float mean = sbuf[0] * (1.f / 512.f);
    const float var  = qbuf[0] * (1.f / 512.f) - mean * mean;
    const float inv  = rsqrtf(var + 1e-5f);
    out[(size_t)row * 512 + t]       = (a - mean) * inv * gamma[t]       + beta[t];
    out[(size_t)row * 512 + t + 256] = (b - mean) * inv * gamma[t + 256] + beta[t + 256];
}

// ---------------------------------------------------------------------------
// Orchestration. Input order: Cp, W1, W2(unused: softmax over size-1 axis
// is identically 1, so scores==1 and W2 cannot affect the output), W3, Wf1,
// bf1, Wf2, bf2, gamma, beta.  Output: [32, 512] fp32.
// ---------------------------------------------------------------------------
extern "C" void kernel_launch(void* const* d_in, const int* in_sizes, int n_in,
                              void* d_out, int out_size, void* d_ws, size_t ws_size,
                              hipStream_t stream) {
    (void)in_sizes; (void)n_in; (void)out_size; (void)ws_size;

    const float* Cp    = (const float*)d_in[0];
    const float* W1    = (const float*)d_in[1];
    const float* W3    = (const float*)d_in[3];
    const float* Wf1   = (const float*)d_in[4];
    const float* bf1   = (const float*)d_in[5];
    const float* Wf2   = (const float*)d_in[6];
    const float* bf2   = (const float*)d_in[7];
    const float* gamma = (const float*)d_in[8];
    const float* beta  = (const float*)d_in[9];
    float* out = (float*)d_out;

    float* ws      = (float*)d_ws;
    float* partial = ws;                       // 256*8*512  = 1,048,576 f
    float* Csum    = partial + 256 * 8 * 512;  // 256*512
    float* S       = Csum    + 256 * 512;      // 256*512
    float* phi     = S       + 256 * 512;      // 32*512
    float* w3      = phi     + 32 * 512;       // 32*512
    float* h1      = w3      + 32 * 512;       // 32*2048
    float* xbuf    = h1      + 32 * 2048;      // 32*512   (~5.7 MB total)

    // 1) Csum[n,h,:] = sum over the 512 rows of head-block h (HBM-bound stage)
    reduce_cp_partial<<<dim3(256, LSPLIT), 128, 0, stream>>>(Cp, partial);
    reduce_cp_combine<<<256, 128, 0, stream>>>(partial, Csum);

    // 2) S = Csum @ W1^T   [256,512]x[512,512]
    wmma_gemm_nt_f32<<<dim3(512 / 16, 256 / 16), 32, 0, stream>>>(
        Csum, W1, nullptr, nullptr, S, 256, 512, 512, 0);

    // 3) phi fold  [32,512]
    fold_phi<<<16384 / 256, 256, 0, stream>>>(S, phi);

    // 4) w3 = phi @ W3^T   [32,512]
    wmma_gemm_nt_f32<<<dim3(512 / 16, 32 / 16), 32, 0, stream>>>(
        phi, W3, nullptr, nullptr, w3, 32, 512, 512, 0);

    // 5) h1 = relu(w3 @ Wf1^T + bf1)   [32,2048]
    wmma_gemm_nt_f32<<<dim3(2048 / 16, 32 / 16), 32, 0, stream>>>(
        w3, Wf1, bf1, nullptr, h1, 32, 2048, 512, 1);

    // 6) x = w3 + (h1 @ Wf2^T + bf2)   [32,512]
    wmma_gemm_nt_f32<<<dim3(512 / 16, 32 / 16), 32, 0, stream>>>(
        h1, Wf2, bf2, w3, xbuf, 32, 512, 2048, 0);

    // 7) LayerNorm rows -> out
    layernorm_rows<<<32, 256, 0, stream>>>(xbuf, gamma, beta, out);
}